// Classifier_75608604279323
// MI455X (gfx1250) — compile-verified
//
#include <hip/hip_runtime.h>
#include <math.h>

// Problem constants (from reference)
#define Bn   1024
#define Kn   30
#define Dn   10
#define Fn   750
#define Tn   10331
#define ISZ  11541            // F + D + 15*K + T
#define DENSE 1210            // F + D + 15*K  (dense prefix of the input vector)
#define XLEN 1216             // DENSE padded to a multiple of 8*4 (WMMA K-step, unroll 8)
#define XHALF (XLEN / 2)      // 608 columns per k-section wave

typedef float v2f __attribute__((ext_vector_type(2)));
typedef float v8f __attribute__((ext_vector_type(8)));

__device__ __forceinline__ float arccosh_ref(float x) {
    x = fmaxf(x, 1.0f + 1e-07f);
    return logf(x + sqrtf(x * x - 1.0f));
}

// One workgroup per batch element b. 128 threads = 4 waves (wave32).
//   wave w: mgroup = w&1 -> neighbor rows 16*mgroup .. 16*mgroup+15
//           ksec   = w>>1 -> column half [ksec*608, ksec*608+608)
// Phase 1: build x_b (1216 floats) in LDS + per-neighbor one-hot/bias terms.
// Phase 2: each wave runs 152 v_wmma_f32_16x16x4_f32 steps
//          (A = 16 gathered W rows x 4 cols, B = x chunk broadcast to 16 cols).
// Phase 3: LDS-combine the two column-half partials, sigmoid -> d_out,
//          deterministic per-block BCE partial -> ws.
__global__ __launch_bounds__(128)
void neigh_logit_kernel(const float* __restrict__ pe,        // [B,D]
                        const int*   __restrict__ idx,       // [B,K]
                        const float* __restrict__ feat,      // [B,F]
                        const float* __restrict__ yoh,       // [B,K]
                        const float* __restrict__ temb,      // [T,D]
                        const float* __restrict__ W,         // [T,ISZ]
                        const float* __restrict__ Wb,        // [T]
                        float*       __restrict__ dist_out,  // [B,K]
                        float*       __restrict__ bce_ws)    // [B]
{
    __shared__ float xs[XLEN];
    __shared__ int   t_sh[32];
    __shared__ float ohb_sh[32];   // bias[t] + unique one-hot gather sum
    __shared__ float part[2][32];  // [ksec][neighbor] partial dot products
    __shared__ float red[32];

    const int b    = blockIdx.x;
    const int tid  = threadIdx.x;
    const int lane = tid & 31;
    const int wave = tid >> 5;

    // ---- Phase 1a: stage dense input prefix ----
    for (int i = tid; i < Fn; i += 128) xs[i] = feat[(size_t)b * Fn + i];
    if (tid < Dn) xs[Fn + tid] = pe[b * Dn + tid];
    if (tid < (XLEN - DENSE)) xs[DENSE + tid] = 0.0f;     // K-dim zero pad
    if (tid < 32) t_sh[tid] = (tid < Kn) ? idx[b * Kn + tid] : 0;
    __syncthreads();

    // ---- Phase 1b: per-neighbor hyperbolic features + one-hot/bias terms ----
    if (tid < Kn) {
        const int t = t_sh[tid];
        float u[Dn], v[Dn];
        #pragma unroll
        for (int d = 0; d < Dn; ++d) { u[d] = xs[Fn + d]; v[d] = temb[(size_t)t * Dn + d]; }
        float uu = 0.f, vv = 0.f, duv = 0.f, dotv = 0.f, pair = 0.f;
        #pragma unroll
        for (int d = 0; d < Dn; ++d) {
            uu += u[d] * u[d];  vv += v[d] * v[d];
            float dd = u[d] - v[d];
            duv  += dd * dd;
            dotv += u[d] * v[d];
            float pp = dd + 1e-06f;
            pair += pp * pp;
        }
        const float uuc = fminf(fmaxf(uu, 0.f), 1.f - 1e-05f);
        const float vvc = fminf(fmaxf(vv, 0.f), 1.f - 1e-05f);
        const float f_poinc = arccosh_ref(1.f + 2.f * duv / ((1.f - uuc) * (1.f - vvc)));
        const float nu = fmaxf(sqrtf(uu), 1e-08f);
        const float nv = fmaxf(sqrtf(vv), 1e-08f);
        const float f_cos  = dotv / (nu * nv);
        const float f_pair = sqrtf(pair);
        const float f_hn   = arccosh_ref(1.f + 2.f * vvc / (1.f - vvc));

        float* xr = &xs[Fn + Dn + 15 * tid];
        #pragma unroll
        for (int d = 0; d < Dn; ++d) xr[d] = v[d];
        xr[10] = f_poinc; xr[11] = f_cos; xr[12] = dotv; xr[13] = f_pair; xr[14] = f_hn;

        // one-hot region: sum over *unique* neighbor indices of W[t, 1210+j], plus bias
        const float* rowoh = W + (size_t)t * ISZ + DENSE;
        float s = Wb[t];
        for (int j = 0; j < Kn; ++j) {
            const int tj = t_sh[j];
            bool first = true;
            for (int j2 = 0; j2 < j; ++j2) first = first && (t_sh[j2] != tj);
            if (first) s += rowoh[tj];
        }
        ohb_sh[tid] = s;
    } else if (tid < 32) {
        ohb_sh[tid] = 0.f;
    }
    __syncthreads();

    // ---- Phase 2: WMMA dot-product accumulation (EXEC all-1s, no divergence) ----
    // 32-bit A 16x4 layout: lanes 0-15 hold {K=0 (vgpr0), K=1 (vgpr1)} of row M=lane;
    // lanes 16-31 hold {K=2, K=3} of row M=lane-16. B mirrors the K striping with
    // every N-column identical (x broadcast), so all 16 D-columns carry the dot.
    const int   mgroup = wave & 1;
    const int   ksec   = wave >> 1;
    const int   half   = lane >> 4;
    const int   mrow   = (mgroup << 4) + (lane & 15);       // padding rows use t=0 (harmless)
    const float* rowp  = W + (size_t)t_sh[mrow] * ISZ;
    const int   kbase  = ksec * XHALF;
    v8f c = {0.f, 0.f, 0.f, 0.f, 0.f, 0.f, 0.f, 0.f};
    #pragma unroll 8
    for (int kc = kbase; kc < kbase + XHALF; kc += 4) {
        const int col = kc + 2 * half;
        v2f a, bb;
        a.x  = rowp[col];  a.y  = rowp[col + 1];            // global_load_b64 (gathered row)
        bb.x = xs[col];    bb.y = xs[col + 1];              // ds_load_b64 (broadcast x)
        c = __builtin_amdgcn_wmma_f32_16x16x4_f32(
                /*neg_a=*/false, a, /*neg_b=*/false, bb,
                /*c_mod=*/(short)0, c, /*reuse_a=*/false, /*reuse_b=*/false);
    }

    // ---- Phase 3a: dump column-half partials from the C layout ----
    // C layout: vgpr v -> M=v on lanes 0-15 (N=lane), M=8+v on lanes 16-31.
    if ((lane & 15) == 0) {
        #pragma unroll
        for (int v = 0; v < 8; ++v) {
            const int m = (mgroup << 4) + (half << 3) + v;
            part[ksec][m] = c[v];
        }
    }
    __syncthreads();

    // ---- Phase 3b: finish logits, sigmoid, BCE ----
    if (tid < Kn) {
        const float logit = (part[0][tid] + part[1][tid]) + ohb_sh[tid];
        dist_out[b * Kn + tid] = 1.f / (1.f + expf(-logit));
        const float y = yoh[b * Kn + tid];
        red[tid] = fmaxf(logit, 0.f) - logit * y + log1pf(expf(-fabsf(logit)));
    }
    __syncthreads();
    if (tid == 0) {
        float s = 0.f;
        #pragma unroll
        for (int m = 0; m < Kn; ++m) s += red[m];   // fixed order -> deterministic
        bce_ws[b] = s;
    }
}

// Deterministic fixed-order final reduction of the per-block BCE partials.
__global__ __launch_bounds__(256)
void loss_reduce_kernel(const float* __restrict__ bce_ws, float* __restrict__ out)
{
    __shared__ float sm[256];
    const int tid = threadIdx.x;
    float s = 0.f;
    for (int i = tid; i < Bn; i += 256) s += bce_ws[i];
    sm[tid] = s;
    __syncthreads();
    for (int off = 128; off > 0; off >>= 1) {
        if (tid < off) sm[tid] += sm[tid + off];
        __syncthreads();
    }
    if (tid == 0) out[Bn * Kn] = sm[0] / (float)(Bn * Kn);
}

extern "C" void kernel_launch(void* const* d_in, const int* in_sizes, int n_in,
                              void* d_out, int out_size, void* d_ws, size_t ws_size,
                              hipStream_t stream)
{
    const float* pe   = (const float*)d_in[0];   // predicted_embeds [B,D]
    const int*   idx  = (const int*)  d_in[1];   // neighbor_indexes [B,K]
    const float* feat = (const float*)d_in[2];   // feature_repre    [B,F]
    const float* yoh  = (const float*)d_in[3];   // one_hot_neighbor_types [B,K]
    const float* temb = (const float*)d_in[4];   // type_emb         [T,D]
    const float* W    = (const float*)d_in[5];   // W_weight         [T,ISZ]
    const float* Wb   = (const float*)d_in[6];   // W_bias           [T]
    float* out = (float*)d_out;                  // [B*K dist | loss]
    float* ws  = (float*)d_ws;                   // [B] BCE partials

    neigh_logit_kernel<<<Bn, 128, 0, stream>>>(pe, idx, feat, yoh, temb, W, Wb, out, ws);
    loss_reduce_kernel<<<1, 256, 0, stream>>>(ws, out);
}